// InfoNCELossForCSC_20263655703050
// MI455X (gfx1250) — compile-verified
//
#include <hip/hip_runtime.h>
#include <cmath>

typedef __attribute__((ext_vector_type(2))) float v2f;
typedef __attribute__((ext_vector_type(8))) float v8f;

#define NEG_K 32
#define WAVES_PER_BLOCK 4

// One wave32 per loss row l.
// WMMA mapping (V_WMMA_F32_16X16X4_F32, exact f32):
//   M = candidate index (16 per tile, 2 tiles for 32 negatives)
//   K = 4-wide slice of H (192 iterations over H=768)
//   B[k][n] = anchor[4j+k] broadcast over all 16 N columns
//   => D[m][n] = dot(negative_m, anchor) for every n (redundant in N, exact).
__global__ __launch_bounds__(WAVES_PER_BLOCK * 32)
void infonce_rows_kernel(const float* __restrict__ hidden,
                         const int* __restrict__ selection,
                         const int* __restrict__ mask,
                         const float* __restrict__ noise,
                         float* __restrict__ row_loss,
                         int S, int H, int loss_size, int valid_size)
{
    const int lane = threadIdx.x & 31;
    const int wid  = threadIdx.x >> 5;
    const int l    = blockIdx.x * WAVES_PER_BLOCK + wid;
    if (l >= loss_size) return;            // wave-uniform: EXEC stays all-ones

    // length[l] = sum(mask[l, :])
    int len = 0;
    for (int i = lane; i < S; i += 32) len += mask[l * S + i];
    #pragma unroll
    for (int off = 16; off > 0; off >>= 1) len += __shfl_xor(len, off, 32);

    const int  sel = selection[l];
    const bool sir = (sel >= 1) && (sel <= len - 1);
    const int  n_valid = sir ? (len - 2) : (len - 1);

    // candidate index for negative k = lane (NEG_K == 32 == wave size)
    const float nz = noise[l * NEG_K + lane];
    int cand = (int)floorf(nz * (float)n_valid) + 1;
    cand = (sir && cand >= sel) ? (cand + 1) : cand;

    const float* base     = hidden + (size_t)l * S * H;
    const float* anchor   = base + (size_t)sel * H;
    const float* positive = hidden + ((size_t)(valid_size + l) * S + (size_t)sel) * H;

    // A-matrix lane mapping: lane L holds candidate (L%16); tile0 = negs 0..15, tile1 = 16..31
    const int c0 = __shfl(cand, lane & 15, 32);
    const int c1 = __shfl(cand, 16 + (lane & 15), 32);
    const float* p0 = base + (size_t)c0 * H;
    const float* p1 = base + (size_t)c1 * H;
    const int koff = (lane >> 4) * 2;      // lanes 0-15: K={0,1}; lanes 16-31: K={2,3}

    v8f acc0 = {0.f, 0.f, 0.f, 0.f, 0.f, 0.f, 0.f, 0.f};
    v8f acc1 = {0.f, 0.f, 0.f, 0.f, 0.f, 0.f, 0.f, 0.f};
    for (int j = 0; j < H / 4; ++j) {
        const int h = j * 4 + koff;
        v2f b, a0, a1;
        b[0]  = anchor[h]; b[1]  = anchor[h + 1];   // B slice broadcast over N
        a0[0] = p0[h];     a0[1] = p0[h + 1];
        a1[0] = p1[h];     a1[1] = p1[h + 1];
        // 8 args: (neg_a, A, neg_b, B, c_mod, C, reuse_a, reuse_b)
        acc0 = __builtin_amdgcn_wmma_f32_16x16x4_f32(false, a0, false, b,
                                                     (short)0, acc0, false, false);
        acc1 = __builtin_amdgcn_wmma_f32_16x16x4_f32(false, a1, false, b,
                                                     (short)0, acc1, false, false);
    }

    // positive logit: plain f32 dot + wave reduction
    float pd = 0.f;
    for (int h = lane; h < H; h += 32) pd = fmaf(anchor[h], positive[h], pd);
    #pragma unroll
    for (int off = 16; off > 0; off >>= 1) pd += __shfl_xor(pd, off, 32);

    // C/D layout: VGPR r holds M=r (lanes 0-15) and M=r+8 (lanes 16-31).
    // So each half-wave redundantly holds 16 of the 32 neg logits:
    //   lanes 0-15 : negs {0..7} in acc0, {16..23} in acc1
    //   lanes 16-31: negs {8..15} in acc0, {24..31} in acc1
    const float invT = 1.0f / 0.07f;
    const float xpos = pd * invT;
    float v[16];
    #pragma unroll
    for (int r = 0; r < 8; ++r) { v[r] = acc0[r] * invT; v[8 + r] = acc1[r] * invT; }

    float mx = xpos;
    #pragma unroll
    for (int r = 0; r < 16; ++r) mx = fmaxf(mx, v[r]);
    mx = fmaxf(mx, __shfl_xor(mx, 16, 32));   // merge the two half-wave maxima

    float s = 0.f;
    #pragma unroll
    for (int r = 0; r < 16; ++r) s += expf(v[r] - mx);
    s += __shfl_xor(s, 16, 32);               // sum over all 32 negatives
    s += expf(xpos - mx);                     // plus the positive term

    if (lane == 0) row_loss[l] = logf(s) + mx - xpos;   // = -log_softmax[0]
}

// Deterministic single-block reduction: mean over B rows, zero rows contribute log(33).
__global__ __launch_bounds__(512)
void infonce_finalize_kernel(const float* __restrict__ row_loss,
                             float* __restrict__ out, int loss_size, int B)
{
    __shared__ float sm[512];
    const int t = threadIdx.x;
    float acc = 0.f;
    for (int i = t; i < loss_size; i += 512) acc += row_loss[i];
    sm[t] = acc;
    __syncthreads();
    #pragma unroll
    for (int stride = 256; stride > 0; stride >>= 1) {
        if (t < stride) sm[t] += sm[t + stride];
        __syncthreads();
    }
    if (t == 0) {
        const float pad = (float)(B - loss_size) * logf((float)(NEG_K + 1));
        out[0] = (sm[0] + pad) / (float)B;
    }
}

extern "C" void kernel_launch(void* const* d_in, const int* in_sizes, int n_in,
                              void* d_out, int out_size, void* d_ws, size_t ws_size,
                              hipStream_t stream) {
    const float* hidden    = (const float*)d_in[0];
    const int*   selection = (const int*)d_in[1];
    const int*   mask      = (const int*)d_in[2];
    const float* noise     = (const float*)d_in[3];
    // d_in[4] is valid_size as a 1-element array; all dims are derivable host-side:
    const int B         = in_sizes[1];
    const int S         = in_sizes[2] / B;
    const int H         = in_sizes[0] / (B * S);
    const int loss_size = in_sizes[3] / NEG_K;
    const int valid_size = B - loss_size;

    float* row_loss = (float*)d_ws;
    float* out = (float*)d_out;

    const int blocks = (loss_size + WAVES_PER_BLOCK - 1) / WAVES_PER_BLOCK;
    infonce_rows_kernel<<<blocks, WAVES_PER_BLOCK * 32, 0, stream>>>(
        hidden, selection, mask, noise, row_loss, S, H, loss_size, valid_size);
    infonce_finalize_kernel<<<1, 512, 0, stream>>>(row_loss, out, loss_size, B);
}